// FrameDockingScoreModel_49435073577009
// MI455X (gfx1250) — compile-verified
//
#include <hip/hip_runtime.h>

typedef float v2f __attribute__((ext_vector_type(2)));
typedef float v8f __attribute__((ext_vector_type(8)));

#define C1_000f 0.24253562503633297f
#define C1_110f 0.14002800840280099f
#define C1_011f 0.3333333333333333f
#define C1_101f 0.3333333333333333f
#define C1_111f 0.23570226039551587f
#define C2_000f 0.10846522890932808f
#define C2_110f 0.06262242910851495f
#define C2_011f 0.14907119849998599f
#define C2_101f 0.14907119849998599f
#define C2_111f 0.10540925533894598f
#define GCOEF   (-0.18f)

// --------------------------------------------------------------------------
// Weight packing: build the two WMMA B-matrices in d_ws.
//  W1p: 20 rows (K) x 32 cols (N).  K=4u+v -> C1_000*w1_000[u][v][w] (w<20),
//       K=16 -> C1_110*w1_110[w].  Col 20 = folded (w2_011 . S) column.
//  W2p: 88 rows (K) x 16 cols (N=9 used). K=4u+v -> C2_000*w2_000[u][v][w],
//       K=80..84 -> C2_110*w2_110[u][w].
// --------------------------------------------------------------------------
__global__ void prep_weights(const float* __restrict__ w1_000,
                             const float* __restrict__ w1_110,
                             const float* __restrict__ w2_011,
                             const float* __restrict__ w2_000,
                             const float* __restrict__ w2_110,
                             float* __restrict__ W1p,
                             float* __restrict__ W2p)
{
  for (int idx = threadIdx.x; idx < 20 * 32; idx += blockDim.x) {
    int k = idx >> 5, w = idx & 31;
    float val = 0.f;
    if (w < 20) {
      if (k < 16)       val = C1_000f * w1_000[k * 20 + w];
      else if (k == 16) val = C1_110f * w1_110[w];
    } else if (w == 20) {
      float s = 0.f;
      if (k < 16)       { for (int j = 0; j < 20; ++j) s += w2_011[j] * C1_000f * w1_000[k * 20 + j]; }
      else if (k == 16) { for (int j = 0; j < 20; ++j) s += w2_011[j] * C1_110f * w1_110[j]; }
      val = s;
    }
    W1p[idx] = val;
  }
  for (int idx = threadIdx.x; idx < 88 * 16; idx += blockDim.x) {
    int k = idx >> 4, w = idx & 15;
    float val = 0.f;
    if (w < 9) {
      if (k < 80)      val = C2_000f * w2_000[k * 9 + w];
      else if (k < 85) val = C2_110f * w2_110[(k - 80) * 9 + w];
    }
    W2p[idx] = val;
  }
}

// --------------------------------------------------------------------------
// Main kernel: one wave handles 16 batch elements per grid-stride iteration.
// Lane L and L+16 both own element (L&15): f32 WMMA A layout needs lane
// halves to supply K=0,1 and K=2,3 of the same row.
// --------------------------------------------------------------------------
__global__ __launch_bounds__(256) void frame_dock_kernel(
    const float* __restrict__ ligand, const float* __restrict__ receptor,
    const float* __restrict__ w1_011, const float* __restrict__ w1_101,
    const float* __restrict__ w1_111, const float* __restrict__ w2_101,
    const float* __restrict__ w2_111,
    const float* __restrict__ W1p, const float* __restrict__ W2p,
    float* __restrict__ out, int nB)
{
  __shared__ float smem[8][16][21];   // per-wave private transpose scratch

  const int lane = threadIdx.x & 31;
  const int wid  = threadIdx.x >> 5;
  const int row  = lane & 15;         // batch row within the 16-row tile / N col
  const int hf   = lane >> 4;         // lane half selects K pair

  // Hoisted small weights (uniform -> scalar loads).
  float A011[20], A101[20], A111[5], B101[20], B111[5];
#pragma unroll
  for (int i = 0; i < 20; ++i) { A011[i] = w1_011[i]; A101[i] = w1_101[i]; B101[i] = w2_101[i]; }
#pragma unroll
  for (int i = 0; i < 5; ++i)  { A111[i] = w1_111[i]; B111[i] = w2_111[i]; }

  // Hoisted WMMA B operands. VGPR pair per K-chunk of 4:
  //  .x = W[k0 + 2*hf][col], .y = W[k0 + 2*hf + 1][col]
  v2f B1t0[5], B1t1[5], B2c[22];
#pragma unroll
  for (int c = 0; c < 5; ++c) {
    int k = 4 * c + 2 * hf;
    B1t0[c].x = W1p[k * 32 + row];        B1t0[c].y = W1p[(k + 1) * 32 + row];
    B1t1[c].x = W1p[k * 32 + 16 + row];   B1t1[c].y = W1p[(k + 1) * 32 + 16 + row];
  }
#pragma unroll
  for (int c = 0; c < 22; ++c) {
    int k = 4 * c + 2 * hf;
    B2c[c].x = W2p[k * 16 + row];         B2c[c].y = W2p[(k + 1) * 16 + row];
  }

  const int  nWaves = (int)((gridDim.x * blockDim.x) >> 5);
  const int  gw     = (int)((blockIdx.x * blockDim.x + threadIdx.x) >> 5);
  const long nB3    = (long)nB * 3;
  const float off1 = 1.6666666f, off2 = 3.3333333f, off3 = 5.0f;

  for (long eb = (long)gw * 16; eb < (long)nB; eb += (long)nWaves * 16) {
    long b  = eb + row;
    long bc = (b < (long)nB) ? b : (long)(nB - 1);

    // prefetch next tile
    {
      long pb = eb + (long)nWaves * 16 + row;
      if (pb < (long)nB) {
        __builtin_prefetch(ligand + pb * 12, 0, 0);
        __builtin_prefetch(receptor + pb * 12, 0, 0);
      }
    }

    const float* lg = ligand + bc * 12;
    const float* rc = receptor + bc * 12;
    // only atoms 1..3 are used by the model
    float e1x = lg[3] - rc[3],  e1y = lg[4] - rc[4],   e1z = lg[5] - rc[5];
    float e2x = lg[6] - rc[6],  e2y = lg[7] - rc[7],   e2z = lg[8] - rc[8];
    float e3x = lg[9] - rc[9],  e3y = lg[10] - rc[10], e3z = lg[11] - rc[11];

    float q1 = e1x * e1x + e1y * e1y + e1z * e1z;
    float q2 = e2x * e2x + e2y * e2y + e2z * e2z;
    float q3 = e3x * e3x + e3y * e3y + e3z * e3z;
    float i1 = rsqrtf(q1), i2 = rsqrtf(q2), i3 = rsqrtf(q3);
    float d1 = q1 * i1, d2 = q2 * i2, d3 = q3 * i3;
    float v1x = e1x * i1, v1y = e1y * i1, v1z = e1z * i1;
    float v2x = e2x * i2, v2y = e2y * i2, v2z = e2z * i2;
    float v3x = e3x * i3, v3y = e3y * i3, v3z = e3z * i3;

    float s1[4], s2[4], s3[4];
    {
      float o[4] = {0.f, off1, off2, off3};
#pragma unroll
      for (int g = 0; g < 4; ++g) {
        float t;
        t = d1 - o[g]; s1[g] = __expf(GCOEF * t * t);
        t = d2 - o[g]; s2[g] = __expf(GCOEF * t * t);
        t = d3 - o[g]; s3[g] = __expf(GCOEF * t * t);
      }
    }

    float dot12 = v1x * v2x + v1y * v2y + v1z * v2z;
    float crx = v1y * v2z - v1z * v2y;
    float cry = v1z * v2x - v1x * v2z;
    float crz = v1x * v2y - v1y * v2x;

    // ---------------- GEMM1: S[16 x 21] = A1[16 x 20] x W1p ----------------
    // A1 row b: K=4u+v -> s1[u]*s2[v], K=16 -> dot12, K=17..19 -> 0
    v8f acc0 = {0, 0, 0, 0, 0, 0, 0, 0};
    v8f acc1 = {0, 0, 0, 0, 0, 0, 0, 0};
    float s2lo = hf ? s2[2] : s2[0];
    float s2hi = hf ? s2[3] : s2[1];
#pragma unroll
    for (int c = 0; c < 4; ++c) {
      v2f A = {s1[c] * s2lo, s1[c] * s2hi};
      acc0 = __builtin_amdgcn_wmma_f32_16x16x4_f32(false, A, false, B1t0[c], (short)0, acc0, false, false);
      acc1 = __builtin_amdgcn_wmma_f32_16x16x4_f32(false, A, false, B1t1[c], (short)0, acc1, false, false);
    }
    {
      v2f A = {hf ? 0.f : dot12, 0.f};
      acc0 = __builtin_amdgcn_wmma_f32_16x16x4_f32(false, A, false, B1t0[4], (short)0, acc0, false, false);
      acc1 = __builtin_amdgcn_wmma_f32_16x16x4_f32(false, A, false, B1t1[4], (short)0, acc1, false, false);
    }

    // ------------- transpose S (D col-striped -> per-lane rows) -------------
#pragma unroll
    for (int r = 0; r < 8; ++r) smem[wid][r + 8 * hf][row] = acc0[r];
    if (row < 5) {
#pragma unroll
      for (int r = 0; r < 8; ++r) smem[wid][r + 8 * hf][16 + row] = acc1[r];
    }
    asm volatile("" ::: "memory");
    float S[20], sdot;
#pragma unroll
    for (int w = 0; w < 20; ++w) S[w] = smem[wid][row][w];
    sdot = smem[wid][row][20];   // folded w2_011 . S
    asm volatile("" ::: "memory");

    // ---------------- per-element V tensor (5 x 3) and Vd3 ----------------
    float a1[5], a2[5];
#pragma unroll
    for (int w = 0; w < 5; ++w) {
      a1[w] = A011[w] * s1[0] + A011[5 + w] * s1[1] + A011[10 + w] * s1[2] + A011[15 + w] * s1[3];
      a2[w] = A101[w] * s2[0] + A101[5 + w] * s2[1] + A101[10 + w] * s2[2] + A101[15 + w] * s2[3];
    }
    float V[5][3], Vd3[5];
#pragma unroll
    for (int w = 0; w < 5; ++w) {
      float c1 = C1_011f * a1[w], c2 = C1_101f * a2[w], c3 = C1_111f * A111[w];
      V[w][0] = c1 * v2x + c2 * v1x + c3 * crx;
      V[w][1] = c1 * v2y + c2 * v1y + c3 * cry;
      V[w][2] = c1 * v2z + c2 * v1z + c3 * crz;
      Vd3[w]  = V[w][0] * v3x + V[w][1] * v3y + V[w][2] * v3z;
    }

    // ---------------- GEMM2: S2[16 x 9] = A2[16 x 88] x W2p ----------------
    // A2 row b: K=4u+v -> S[u]*s3[v], K=80..84 -> Vd3[u], K=85..87 -> 0
    v8f acc2 = {0, 0, 0, 0, 0, 0, 0, 0};
    float s3lo = hf ? s3[2] : s3[0];
    float s3hi = hf ? s3[3] : s3[1];
#pragma unroll
    for (int c = 0; c < 20; ++c) {
      v2f A = {S[c] * s3lo, S[c] * s3hi};
      acc2 = __builtin_amdgcn_wmma_f32_16x16x4_f32(false, A, false, B2c[c], (short)0, acc2, false, false);
    }
    {
      v2f A;
      A.x = hf ? Vd3[2] : Vd3[0];
      A.y = hf ? Vd3[3] : Vd3[1];
      acc2 = __builtin_amdgcn_wmma_f32_16x16x4_f32(false, A, false, B2c[20], (short)0, acc2, false, false);
    }
    {
      v2f A;
      A.x = hf ? 0.f : Vd3[4];
      A.y = 0.f;
      acc2 = __builtin_amdgcn_wmma_f32_16x16x4_f32(false, A, false, B2c[21], (short)0, acc2, false, false);
    }

    // ------------------------- transpose S2 back -------------------------
    asm volatile("" ::: "memory");
#pragma unroll
    for (int r = 0; r < 8; ++r) smem[wid][r + 8 * hf][row] = acc2[r];
    asm volatile("" ::: "memory");
    float S2[9];
#pragma unroll
    for (int w = 0; w < 9; ++w) S2[w] = smem[wid][row][w];

    // ------------------------------- V2 ----------------------------------
    float g[5];
#pragma unroll
    for (int u = 0; u < 5; ++u)
      g[u] = B101[u * 4 + 0] * s3[0] + B101[u * 4 + 1] * s3[1] +
             B101[u * 4 + 2] * s3[2] + B101[u * 4 + 3] * s3[3];
    float sVx = 0.f, sVy = 0.f, sVz = 0.f;
    float Vwx = 0.f, Vwy = 0.f, Vwz = 0.f;
#pragma unroll
    for (int u = 0; u < 5; ++u) {
      sVx += g[u] * V[u][0];  sVy += g[u] * V[u][1];  sVz += g[u] * V[u][2];
      Vwx += B111[u] * V[u][0]; Vwy += B111[u] * V[u][1]; Vwz += B111[u] * V[u][2];
    }
    float cxx = Vwy * v3z - Vwz * v3y;
    float cxy = Vwz * v3x - Vwx * v3z;
    float cxz = Vwx * v3y - Vwy * v3x;
    float sc = C2_011f * sdot;
    float V2x = sc * v3x + C2_101f * sVx + C2_111f * cxx;
    float V2y = sc * v3y + C2_101f * sVy + C2_111f * cxy;
    float V2z = sc * v3z + C2_101f * sVz + C2_111f * cxz;

    // ---------------- stores: 4 segments of (B,3), lane half 0 only --------
    if (hf == 0 && b < (long)nB) {
      long o = b * 3;
      out[o + 0] = S2[0]; out[o + 1] = S2[1]; out[o + 2] = S2[2];
      out[nB3 + o + 0] = S2[3]; out[nB3 + o + 1] = S2[4]; out[nB3 + o + 2] = S2[5];
      out[2 * nB3 + o + 0] = S2[6]; out[2 * nB3 + o + 1] = S2[7]; out[2 * nB3 + o + 2] = S2[8];
      out[3 * nB3 + o + 0] = V2x; out[3 * nB3 + o + 1] = V2y; out[3 * nB3 + o + 2] = V2z;
    }
  }
}

extern "C" void kernel_launch(void* const* d_in, const int* in_sizes, int n_in,
                              void* d_out, int out_size, void* d_ws, size_t ws_size,
                              hipStream_t stream) {
  const float* ligand   = (const float*)d_in[0];
  const float* receptor = (const float*)d_in[1];
  const float* w1_000   = (const float*)d_in[2];
  const float* w1_110   = (const float*)d_in[3];
  const float* w1_011   = (const float*)d_in[4];
  const float* w1_101   = (const float*)d_in[5];
  const float* w1_111   = (const float*)d_in[6];
  const float* w2_000   = (const float*)d_in[7];
  const float* w2_110   = (const float*)d_in[8];
  const float* w2_011   = (const float*)d_in[9];
  const float* w2_101   = (const float*)d_in[10];
  const float* w2_111   = (const float*)d_in[11];

  int nB = in_sizes[0] / 12;
  float* W1p = (float*)d_ws;        // 20*32 = 640 floats
  float* W2p = W1p + 640;           // 88*16 = 1408 floats

  prep_weights<<<1, 256, 0, stream>>>(w1_000, w1_110, w2_011, w2_000, w2_110, W1p, W2p);

  int wavesNeeded = (nB + 15) / 16;
  int blocks = (wavesNeeded + 7) / 8;
  if (blocks > 2048) blocks = 2048;
  if (blocks < 1) blocks = 1;
  frame_dock_kernel<<<blocks, 256, 0, stream>>>(
      ligand, receptor, w1_011, w1_101, w1_111, w2_101, w2_111,
      W1p, W2p, (float*)d_out, nB);
}